// DifferentiableAstar_64398739636891
// MI455X (gfx1250) — compile-verified
//
#include <hip/hip_runtime.h>

// Differentiable A* (B=16, H=W=32, N=1024) for MI455X / gfx1250.
//
// Single 512-thread workgroup = 16 wave32s; wave b owns batch b, lane l owns
// row l (32 cells in registers). Per step: unrolled f=0.5(g+h), exp, argmax
// via 5 shfl_xor butterflies (wave32), relax <=8 neighbors around the
// selected cell, 2 barriers for the cross-batch "done" check. cost[] is
// staged into LDS with CDNA5 async global->LDS loads (ASYNCcnt), overlapped
// with the init phase (heuristic + goal-index reduction), so the per-step
// data-dependent cost[s] read is LDS-latency. parents dumped to LDS for the
// sequential backtrack. No WMMA on purpose: zero GEMM content in the
// latency-bound sequential scan.

#define BATCH 16
#define HH 32
#define WW 32
#define NN (HH * WW)
#define INV_SQRT_W 0.17677669529663687f  // 1/sqrt(32)

__global__ __launch_bounds__(512) void astar_gfx1250_kernel(
    const float* __restrict__ cost,
    const float* __restrict__ start,
    const float* __restrict__ goal,
    const float* __restrict__ obst,
    float* __restrict__ out) {
  __shared__ float sh_cost[BATCH * NN];             // 64 KB, async-staged
  __shared__ unsigned short sh_par[BATCH][NN];      // 32 KB, for backtrack
  __shared__ unsigned int sh_pathbits[BATCH][32];   // 2 KB packed path
  __shared__ int sh_flag[BATCH];

  const int tid = threadIdx.x;
  const int w = tid >> 5;   // wave id == batch id
  const int l = tid & 31;   // lane id == row id

  // ---- Issue async global->LDS staging of cost[] (CDNA5 ASYNCcnt path).
  // Do NOT wait yet: overlap the DMA with the whole init phase below.
  {
    unsigned ldsbase = (unsigned)(unsigned long long)(&sh_cost[0]);
    for (int i = 0; i < 8; ++i) {
      int v4 = tid + i * 512;                 // 4096 16-byte chunks total
      const float* gp = cost + (size_t)v4 * 4;
      unsigned ldsoff = ldsbase + (unsigned)v4 * 16u;
      asm volatile("global_load_async_to_lds_b128 %0, %1, off"
                   :: "v"(ldsoff), "v"(gp) : "memory");
    }
  }

  // ---- Per-lane register state: 32 cells of row l of batch w ----
  float g[WW], h[WW];
  int par[WW];
  unsigned open_m = 0, hist_m = 0, obst_m = 0;

  const int rowbase = w * NN + l * WW;
  int gflat = 0x7fffffff;
#pragma unroll
  for (int k = 0; k < WW; ++k) {
    if (start[rowbase + k] > 0.5f) open_m |= (1u << k);
    if (obst[rowbase + k] != 0.0f) obst_m |= (1u << k);
    if (goal[rowbase + k] > 0.5f) { int fl = l * WW + k; if (fl < gflat) gflat = fl; }
    g[k] = 0.0f;
  }
  // goal index (one-hot; min over wave == jnp.argmax first-max rule)
  for (int off = 16; off; off >>= 1) {
    int o = __shfl_xor(gflat, off, 32);
    if (o < gflat) gflat = o;
  }
  const int rg = gflat >> 5, cg = gflat & 31;
#pragma unroll
  for (int k = 0; k < WW; ++k) {
    float dr = (float)(l - rg), dc = (float)(k - cg);
    // h = Chebyshev + TB * Euclidean  (== |d|.sum - |d|.min + TB*euc)
    h[k] = fmaxf(fabsf(dr), fabsf(dc)) + 0.001f * sqrtf(dr * dr + dc * dc);
    par[k] = gflat;  // parents0 = goal index everywhere
  }

  // Now drain this wave's async copies; barrier makes all waves' LDS visible.
  asm volatile("s_wait_asynccnt 0" ::: "memory");
  __syncthreads();

  // ---- Sequential A* scan (up to N steps, freezes when all solved) ----
  int tfin = 0;
  for (int t = 0; t < NN; ++t) {
    tfin = t;

    // argmax of exp(-f/sqrt(W)) * open, tie -> lowest flat index
    float bv = -1.0f, bg = 0.0f;
    int bi = 0;
#pragma unroll
    for (int k = 0; k < WW; ++k) {
      float f = 0.5f * (g[k] + h[k]);
      float openf = (open_m >> k & 1u) ? 1.0f : 0.0f;
      float val = __expf(-f * INV_SQRT_W) * openf;
      if (val > bv) { bv = val; bi = l * WW + k; bg = g[k]; }
    }
    for (int off = 16; off; off >>= 1) {   // wave32 butterfly
      float ov = __shfl_xor(bv, off, 32);
      int oi = __shfl_xor(bi, off, 32);
      float og = __shfl_xor(bg, off, 32);
      if (ov > bv || (ov == bv && oi < bi)) { bv = ov; bi = oi; bg = og; }
    }
    const int s = bi, rs = s >> 5, cs = s & 31;
    const bool solved = (s == gflat);  // dist_to_goal = goal[s]

    // hist[s]=1 ; open[s]-=is_unsolved (s is excluded from neighbors)
    if (l == rs) {
      hist_m |= (1u << cs);
      if (!solved) open_m &= ~(1u << cs);
    }
    const float g2val = bg + sh_cost[w * NN + s];  // LDS-latency dependent read

    // relax the 8-neighborhood of s (SAME padding via row/col bounds)
    if (l >= rs - 1 && l <= rs + 1) {
#pragma unroll
      for (int k = 0; k < WW; ++k) {
        bool isnb = (k >= cs - 1) && (k <= cs + 1) && !((l == rs) && (k == cs));
        if (isnb) {
          bool ob = (open_m >> k) & 1u;
          bool hb = (hist_m >> k) & 1u;
          bool obb = (obst_m >> k) & 1u;
          bool upd = obb && ((!ob && !hb) || (ob && (g[k] > g2val)));
          if (upd) { g[k] = g2val; open_m |= (1u << k); par[k] = s; }
        }
      }
    }

    // done = all 16 batches selected their goal this step -> freeze & exit
    if (l == 0) sh_flag[w] = solved ? 1 : 0;
    __syncthreads();
    bool alls = true;
    for (int i = 0; i < BATCH; ++i) alls = alls && (sh_flag[i] != 0);
    __syncthreads();
    if (alls) break;  // uniform across block; t_fin = this t
  }

  // ---- Output 1: hist ----
#pragma unroll
  for (int k = 0; k < WW; ++k)
    out[w * NN + l * WW + k] = (hist_m >> k & 1u) ? 1.0f : 0.0f;

  // ---- Backtrack: parents -> LDS, one lane walks the chain ----
#pragma unroll
  for (int k = 0; k < WW; ++k)
    sh_par[w][l * WW + k] = (unsigned short)par[k];
  sh_pathbits[w][l] = ((gflat >> 5) == l) ? (1u << (gflat & 31)) : 0u;  // goal one-hot
  __syncthreads();
  if (l == 0) {
    int loc = sh_par[w][gflat];  // loc0 = parents[goal_idx]
    for (int i = 0; i < tfin; ++i) {
      sh_pathbits[w][loc >> 5] |= (1u << (loc & 31));
      loc = sh_par[w][loc];
    }
  }
  __syncthreads();

  // ---- Output 2: path ----
  unsigned pw = sh_pathbits[w][l];
#pragma unroll
  for (int k = 0; k < WW; ++k)
    out[BATCH * NN + w * NN + l * WW + k] = (pw >> k & 1u) ? 1.0f : 0.0f;
}

extern "C" void kernel_launch(void* const* d_in, const int* in_sizes, int n_in,
                              void* d_out, int out_size, void* d_ws, size_t ws_size,
                              hipStream_t stream) {
  const float* cost  = (const float*)d_in[0];   // cost_maps      (16,1,32,32)
  const float* start = (const float*)d_in[1];   // start_maps
  const float* goal  = (const float*)d_in[2];   // goal_maps
  // d_in[3] = heuristic_maps: unused by the reference (_heuristic_dist is used)
  const float* obst  = (const float*)d_in[4];   // obstacles_maps
  float* out = (float*)d_out;                   // [hist | path] = 2*16*1024 floats

  astar_gfx1250_kernel<<<1, 512, 0, stream>>>(cost, start, goal, obst, out);
}